// TimeGCN_GRU_14508399526046
// MI455X (gfx1250) — compile-verified
//
#include <hip/hip_runtime.h>
#include <hip/hip_bf16.h>

// Problem constants (from the reference)
#define B_DIM   4
#define L_DIM   30
#define N_NODES 4096
#define F_DIM   16
#define G_DIM   32
#define H_DIM   64
#define M_ROWS  (B_DIM * N_NODES)   // 16384 independent GRU rows

typedef float v2f __attribute__((ext_vector_type(2)));
typedef float v8f __attribute__((ext_vector_type(8)));
typedef __attribute__((ext_vector_type(4))) int i32x4;

// ---------------------------------------------------------------------------
// CDNA5 async global->LDS copy (ASYNCcnt-tracked), guarded for portability.
// Builtin signature (from hipcc diagnostic): first param is a non-const
// global-AS pointer to int4; we pass (global src, lds dst, imm off, imm cpol).
// ---------------------------------------------------------------------------
#if defined(__HIP_DEVICE_COMPILE__) &&                                        \
    __has_builtin(__builtin_amdgcn_global_load_async_to_lds_b128) &&          \
    __has_builtin(__builtin_amdgcn_s_wait_asynccnt)
#define HAVE_ASYNC_COPY 1
#else
#define HAVE_ASYNC_COPY 0
#endif

#if HAVE_ASYNC_COPY
typedef __attribute__((address_space(1))) i32x4* gp128_t;
typedef __attribute__((address_space(3))) i32x4* lp128_t;
__device__ __forceinline__ void async_cp16(const void* g, void* l) {
  __builtin_amdgcn_global_load_async_to_lds_b128(
      (gp128_t)(void*)g, (lp128_t)l, 0, 0);
}
#define ASYNC_WAIT(n) __builtin_amdgcn_s_wait_asynccnt(n)
#else
__device__ __forceinline__ void async_cp16(const void* g, void* l) {
  const float4 v = *(const float4*)g;
  float* d = (float*)l;
  d[0] = v.x; d[1] = v.y; d[2] = v.z; d[3] = v.w;
}
#define ASYNC_WAIT(n) ((void)0)
#endif

// V_WMMA_F32_16X16X4_F32: D(16x16,f32) = A(16x4,f32) x B(4x16,f32) + C
// Fragment layout (ISA 7.12.2):
//   A: lane m = lane&15 holds row m; VGPR0 = K=(lane>>4)*2, VGPR1 = K+1
//   B: lane col = lane&15; VGPR0 = row K=(lane>>4)*2, VGPR1 = row K+1
//   C/D: lane col = lane&15; VGPR r holds row (r + (lane>>4)*8)
__device__ __forceinline__ v8f wmma4(v2f a, v2f b, v8f c) {
  return __builtin_amdgcn_wmma_f32_16x16x4_f32(
      /*neg_a=*/false, a, /*neg_b=*/false, b,
      /*c_mod=*/(short)0, c, /*reuse_a=*/false, /*reuse_b=*/false);
}

__device__ __forceinline__ float sigmoidf_(float x) {
  return 1.0f / (1.0f + __expf(-x));
}

// LDS pitches: multiples of 4 floats (16B) so async b128 LDS writes are
// aligned; 36*m mod 64 and 20*m mod 64 give conflict-free A-frag reads.
#define PA 36   // pitch of A tile rows (floats)
#define PX 20   // pitch of X tile rows (floats)

// ---------------------------------------------------------------------------
// Kernel 1: per (b,l): Y = A[4096x4096] @ X[4096x16]; H = relu(Y @ Wg^T + bg)
// One wave per 64-row macro-tile (4 WMMA M-tiles -> B-frag reused 4x).
// A/X K-chunks double-buffered through LDS with async global->LDS copies.
// hbuf layout: [L][B*N][G]  (each GRU timestep's x-matrix is contiguous)
// ---------------------------------------------------------------------------
__global__ __launch_bounds__(32)
void diffuse_gcn_kernel(const float* __restrict__ A,
                        const float* __restrict__ x_seq,
                        const float* __restrict__ gcn_W,
                        const float* __restrict__ gcn_b,
                        float* __restrict__ hbuf) {
  const int lane = threadIdx.x;        // 0..31
  const int col  = lane & 15;
  const int hi   = lane >> 4;
  const int ko   = hi * 2;

  const int m0 = blockIdx.x * 64;      // 64-row macro tile
  const int bl = blockIdx.y;           // 0..B*L-1
  const int b  = bl & (B_DIM - 1);
  const int l  = bl >> 2;

  const float* X = x_seq + ((size_t)(b * L_DIM + l) * N_NODES) * F_DIM; // [4096,16]

  __shared__ float sA[2][64][PA];      // A tile 64x32, double buffered
  __shared__ float sX[2][32][PX];      // X tile 32x16, double buffered
  __shared__ float sY[16][17];         // Y tile staging for 2nd GEMM
  __shared__ float sW[16][33];         // Wg^T : [F=16][G=32]

  // load gcn_W^T (32x16 -> 16x32), coalesced
  for (int i = 0; i < 16; ++i) {
    int e = i * 32 + lane;             // 0..511
    sW[e & 15][e >> 4] = gcn_W[(e >> 4) * F_DIM + (e & 15)];
  }

  // 20 async b128 ops per K-chunk: 16 for the A tile, 4 for the X tile
  auto issue = [&](int buf, int k0) {
#pragma unroll
    for (int j = 0; j < 16; ++j) {
      int t  = j * 32 + lane;          // 512 float4 slots, 8 per row
      int r  = t >> 3;
      int c4 = t & 7;
      async_cp16(A + (size_t)(m0 + r) * N_NODES + k0 + c4 * 4,
                 &sA[buf][r][c4 * 4]);
    }
#pragma unroll
    for (int j = 0; j < 4; ++j) {
      int t  = j * 32 + lane;          // 128 float4 slots, 4 per row
      int r  = t >> 2;
      int c4 = t & 3;
      async_cp16(X + (size_t)k0 * F_DIM + t * 4, &sX[buf][r][c4 * 4]);
    }
  };

  v8f acc[4] = {};
  issue(0, 0);
  for (int k0 = 0; k0 < N_NODES; k0 += 32) {
    const int buf = (k0 >> 5) & 1;
    if (k0 + 32 < N_NODES) {
      issue(buf ^ 1, k0 + 32);
      ASYNC_WAIT(20);                  // chunk `buf` landed; next 20 in flight
    } else {
      ASYNC_WAIT(0);
    }
#pragma unroll
    for (int kk = 0; kk < 32; kk += 4) {
      v2f bf = { sX[buf][kk + ko][col], sX[buf][kk + ko + 1][col] };
#pragma unroll
      for (int mt = 0; mt < 4; ++mt) { // B-frag reused across 4 M-tiles
        v2f af = { sA[buf][mt * 16 + col][kk + ko],
                   sA[buf][mt * 16 + col][kk + ko + 1] };
        acc[mt] = wmma4(af, bf, acc[mt]);
      }
    }
  }

  // Fused GCN: H = relu(Y[16x16] @ Wg^T[16x32] + bg) for each M-tile.
#pragma unroll
  for (int mt = 0; mt < 4; ++mt) {
#pragma unroll
    for (int r = 0; r < 8; ++r) sY[r + hi * 8][col] = acc[mt][r];
    // same-wave DS ops are in-order: stores above precede loads below
    const size_t outbase =
        ((size_t)l * M_ROWS + (size_t)b * N_NODES + m0 + mt * 16) * G_DIM;
#pragma unroll
    for (int nh = 0; nh < 2; ++nh) {
      const float bb = gcn_b[nh * 16 + col];
      v8f c2;
#pragma unroll
      for (int r = 0; r < 8; ++r) c2[r] = bb;
#pragma unroll
      for (int kk = 0; kk < 16; kk += 4) {
        v2f af = { sY[col][kk + ko], sY[col][kk + ko + 1] };
        v2f bf = { sW[kk + ko][nh * 16 + col], sW[kk + ko + 1][nh * 16 + col] };
        c2 = wmma4(af, bf, c2);
      }
#pragma unroll
      for (int r = 0; r < 8; ++r) {
        float v = c2[r];
        v = v > 0.0f ? v : 0.0f;
        hbuf[outbase + (size_t)(r + hi * 8) * G_DIM + nh * 16 + col] = v;
      }
    }
  }
}

// ---------------------------------------------------------------------------
// Kernel 2: GRU over 30 steps + head. Rows independent -> each wave owns a
// 16-row tile for all timesteps; h state lives in per-wave LDS. Next step's
// x-tile is async-prefetched into the other LDS buffer during compute.
// Weights (72KB) stay hot in L2 across all 1024 waves.
// ---------------------------------------------------------------------------
__global__ __launch_bounds__(128)
void gru_head_kernel(const float* __restrict__ hbuf,
                     const float* __restrict__ W_ih,
                     const float* __restrict__ W_hh,
                     const float* __restrict__ b_ih,
                     const float* __restrict__ b_hh,
                     const float* __restrict__ head_W,
                     const float* __restrict__ head_b,
                     float* __restrict__ out) {
  const int lane = threadIdx.x & 31;
  const int w    = threadIdx.x >> 5;   // wave in block (0..3)
  const int col  = lane & 15;
  const int hi   = lane >> 4;
  const int ko   = hi * 2;
  const int m0   = (blockIdx.x * 4 + w) * 16;

  __shared__ float sH[4][16][65];      // per-wave h state [16 x 64] (+pad)
  __shared__ float sX[4][2][16][PX];   // per-wave x_t tile, double buffered

  // h0 = 0
  for (int i = 0; i < 32; ++i) {
    int e = i * 32 + lane;             // 0..1023
    sH[w][e >> 6][e & 63] = 0.0f;
  }

  auto issueX = [&](int buf, int t) {  // 4 async b128 ops (16x32 tile)
    const float* xt = hbuf + ((size_t)t * M_ROWS + m0) * G_DIM;
#pragma unroll
    for (int j = 0; j < 4; ++j) {
      int q  = j * 32 + lane;          // float4 slot, 8 per row
      int r  = q >> 3;
      int c4 = q & 7;
      async_cp16(xt + q * 4, &sX[w][buf][r][c4 * 4]);
    }
  };

  issueX(0, 0);
  for (int t = 0; t < L_DIM; ++t) {
    const int buf = t & 1;
    if (t + 1 < L_DIM) {
      issueX(buf ^ 1, t + 1);
      ASYNC_WAIT(4);                   // this step's tile landed
    } else {
      ASYNC_WAIT(0);
    }

    v8f grz[8];    // gates r,z: output cols 0..127 of the 192
    v8f gin[4];    // i_n: cols 128..191 (input path)
    v8f ghn[4];    // h_n: cols 128..191 (hidden path)
#pragma unroll
    for (int c = 0; c < 8; ++c) {
      const float bb = b_ih[c * 16 + col] + b_hh[c * 16 + col];
#pragma unroll
      for (int r = 0; r < 8; ++r) grz[c][r] = bb;
    }
#pragma unroll
    for (int c = 0; c < 4; ++c) {
      const float bi = b_ih[128 + c * 16 + col];
      const float bh = b_hh[128 + c * 16 + col];
#pragma unroll
      for (int r = 0; r < 8; ++r) { gin[c][r] = bi; ghn[c][r] = bh; }
    }

    // r/z gates: x[16x32] @ W_ih^T + h[16x64] @ W_hh^T (accumulator chained)
#pragma unroll
    for (int c = 0; c < 8; ++c) {
      const int o = c * 16 + col;      // output column this lane covers
#pragma unroll
      for (int kk = 0; kk < G_DIM; kk += 4) {
        v2f af = { sX[w][buf][col][kk + ko], sX[w][buf][col][kk + ko + 1] };
        v2f bf = { W_ih[(size_t)o * G_DIM + kk + ko],
                   W_ih[(size_t)o * G_DIM + kk + ko + 1] };
        grz[c] = wmma4(af, bf, grz[c]);
      }
#pragma unroll
      for (int kk = 0; kk < H_DIM; kk += 4) {
        v2f af = { sH[w][col][kk + ko], sH[w][col][kk + ko + 1] };
        v2f bf = { W_hh[(size_t)o * H_DIM + kk + ko],
                   W_hh[(size_t)o * H_DIM + kk + ko + 1] };
        grz[c] = wmma4(af, bf, grz[c]);
      }
    }
    // i_n and h_n kept separate: n = tanh(i_n + r * h_n)
#pragma unroll
    for (int c = 0; c < 4; ++c) {
      const int o = 128 + c * 16 + col;
#pragma unroll
      for (int kk = 0; kk < G_DIM; kk += 4) {
        v2f af = { sX[w][buf][col][kk + ko], sX[w][buf][col][kk + ko + 1] };
        v2f bf = { W_ih[(size_t)o * G_DIM + kk + ko],
                   W_ih[(size_t)o * G_DIM + kk + ko + 1] };
        gin[c] = wmma4(af, bf, gin[c]);
      }
#pragma unroll
      for (int kk = 0; kk < H_DIM; kk += 4) {
        v2f af = { sH[w][col][kk + ko], sH[w][col][kk + ko + 1] };
        v2f bf = { W_hh[(size_t)o * H_DIM + kk + ko],
                   W_hh[(size_t)o * H_DIM + kk + ko + 1] };
        ghn[c] = wmma4(af, bf, ghn[c]);
      }
    }

    // gate math in C/D layout; update h in LDS (same-wave DS is in-order)
#pragma unroll
    for (int c = 0; c < 4; ++c) {
#pragma unroll
      for (int r = 0; r < 8; ++r) {
        const float rg = sigmoidf_(grz[c][r]);
        const float zg = sigmoidf_(grz[c + 4][r]);
        const float ng = tanhf(gin[c][r] + rg * ghn[c][r]);
        const int row = r + hi * 8;
        const int j   = c * 16 + col;
        const float ho = sH[w][row][j];
        sH[w][row][j] = (1.0f - zg) * ng + zg * ho;
      }
    }
  }

  // head: logits[m] = h_final[m] . head_W + head_b
  if (lane < 16) {
    float s = head_b[0];
#pragma unroll
    for (int j = 0; j < H_DIM; ++j) s += sH[w][lane][j] * head_W[j];
    out[m0 + lane] = s;
  }
}

// ---------------------------------------------------------------------------
// Launch. d_ws needs L*M*G floats = 62,914,560 bytes for the h buffer.
// ---------------------------------------------------------------------------
extern "C" void kernel_launch(void* const* d_in, const int* in_sizes, int n_in,
                              void* d_out, int out_size, void* d_ws, size_t ws_size,
                              hipStream_t stream) {
  const float* x_seq  = (const float*)d_in[0];
  const float* A      = (const float*)d_in[1];
  const float* gcn_W  = (const float*)d_in[2];
  const float* gcn_b  = (const float*)d_in[3];
  const float* W_ih   = (const float*)d_in[4];
  const float* W_hh   = (const float*)d_in[5];
  const float* b_ih   = (const float*)d_in[6];
  const float* b_hh   = (const float*)d_in[7];
  const float* head_W = (const float*)d_in[8];
  const float* head_b = (const float*)d_in[9];
  float* out  = (float*)d_out;
  float* hbuf = (float*)d_ws;   // [L][B*N][G] fp32

  dim3 g1(N_NODES / 64, B_DIM * L_DIM);
  diffuse_gcn_kernel<<<g1, 32, 0, stream>>>(A, x_seq, gcn_W, gcn_b, hbuf);

  gru_head_kernel<<<M_ROWS / 64, 128, 0, stream>>>(
      hbuf, W_ih, W_hh, b_ih, b_hh, head_W, head_b, out);
}